// InstantPolicyAgent_44693429682828
// MI455X (gfx1250) — compile-verified
//
#include <hip/hip_runtime.h>

typedef float v2f __attribute__((ext_vector_type(2)));
typedef float v8f __attribute__((ext_vector_type(8)));

#define F_IN  64
#define HDIM  64
#define F_E   16

// ---------------------------------------------------------------------------
// Kernel 1: per-node-type projections.
//   h1 = x @ W1[node_type] + b1[node_type]  -> d_out   (base term of output)
//   h2 = x @ W2[node_type] + b2[node_type]  -> d_ws    (gathered by edges)
//
// W1+W2 (both node types, 64 KB) are staged once per block into dynamic LDS
// in ROW-PAIR-INTERLEAVED layout:
//     sP[h][t][r][col] = { W[2r][col], W[2r+1][col] }   (float2)
// so each WMMA B-fragment {W[kk][col], W[kk+1][col]} is ONE contiguous
// ds_load_b64 straight into an even-aligned VGPR pair (no operand-assembly
// v_movs). Each wave32 computes a 16-node x 64-col tile with
// V_WMMA_F32_16X16X4_F32; both node types accumulated, selected per node.
// ---------------------------------------------------------------------------
__global__ __launch_bounds__(256)
void node_proj_kernel(const float* __restrict__ x,
                      const int*   __restrict__ node_type,
                      const float* __restrict__ W1, const float* __restrict__ b1,
                      const float* __restrict__ W2, const float* __restrict__ b2,
                      float* __restrict__ h1_out, float* __restrict__ h2_out,
                      int N)
{
    extern __shared__ float smem[];           // 16384 floats = 64 KB
    v2f* s2 = (v2f*)smem;                     // 8192 float2: [2][2][32][64]

    // Cooperative interleaving stage: h=0 -> W1, h=1 -> W2.
    // pair index p = t*2048 + r*64 + col  (t:type, r:row-pair, col:column)
    {
        const float* gsrc[2] = { W1, W2 };
        #pragma unroll
        for (int h = 0; h < 2; ++h) {
            const float* g = gsrc[h];
            #pragma unroll
            for (int i = 0; i < 16; ++i) {
                const int p   = threadIdx.x + 256 * i;   // 0..4095
                const int col = p & 63;
                const int r   = (p >> 6) & 31;
                const int t   = p >> 11;
                const float* src = g + (size_t)t * 4096 + (size_t)(2 * r) * 64 + col;
                v2f pr;
                pr.x = src[0];
                pr.y = src[64];
                s2[h * 4096 + p] = pr;                   // ds_store_b64
            }
        }
    }
    __syncthreads();

    const int lane    = threadIdx.x & 31;
    const int wave    = threadIdx.x >> 5;
    const int rowbase = (blockIdx.x * 8 + wave) * 16;
    if (rowbase >= N) return;                 // N % 16 == 0, no partial tiles

    const int m     = lane & 15;              // A: row-in-tile, B/C: col-in-subtile
    const int khalf = lane >> 4;              // 0 -> K{0,1}, 1 -> K{2,3} per step

    // A fragments: x[rowbase+m][0..63], 16 K-steps of K=4
    v2f a[16];
    const float* xrow = x + (size_t)(rowbase + m) * F_IN;
    #pragma unroll
    for (int k = 0; k < 16; ++k) {
        const int kk = 4 * k + 2 * khalf;
        a[k].x = xrow[kk];
        a[k].y = xrow[kk + 1];
    }

    // node types for the 8 M-rows (M = v + 8*khalf) this lane's C covers
    int nt[8];
    #pragma unroll
    for (int v = 0; v < 8; ++v)
        nt[v] = node_type[rowbase + v + 8 * khalf];

    const float* bvec[2] = { b1, b2 };
    float*       outp[2] = { h1_out, h2_out };

    #pragma unroll
    for (int h = 0; h < 2; ++h) {
        const v2f*   sWp = s2 + h * 4096;     // [2][32][64] float2 in LDS
        const float* bb  = bvec[h];           // [2][64] in global (tiny, cached)
        float*       op  = outp[h];
        #pragma unroll
        for (int ntile = 0; ntile < 4; ++ntile) {
            const int col = ntile * 16 + m;
            v8f acc0 = {};                    // node_type 0
            v8f acc1 = {};                    // node_type 1
            #pragma unroll
            for (int k = 0; k < 16; ++k) {
                const int r = 2 * k + khalf;  // row-pair index = kk/2
                const v2f bt0 = sWp[r * 64 + col];            // ds_load_b64
                const v2f bt1 = sWp[2048 + r * 64 + col];     // ds_load_b64
                acc0 = __builtin_amdgcn_wmma_f32_16x16x4_f32(
                           false, a[k], false, bt0, (short)0, acc0, false, false);
                acc1 = __builtin_amdgcn_wmma_f32_16x16x4_f32(
                           false, a[k], false, bt1, (short)0, acc1, false, false);
            }
            #pragma unroll
            for (int v = 0; v < 8; ++v) {
                const int   node = rowbase + v + 8 * khalf;
                const int   t    = nt[v];
                const float val  = (t == 0 ? acc0[v] : acc1[v]) + bb[t * HDIM + col];
                op[(size_t)node * HDIM + col] = val;
            }
        }
    }
}

// ---------------------------------------------------------------------------
// Kernel 2: per-edge-type projection + message scatter (dominant kernel).
//   w5e = edge_feat @ W5[edge_type] + b5[edge_type]
//   out[dst] += h2[src] + w5e        (softmax over singleton axis == 1.0)
// Persistent grid-stride waves: all W5 B-fragments (3 types x 4 col-tiles x
// 4 K-steps = 96 VGPRs) and biases are loaded ONCE per wave and live in
// registers; the per-tile work is only the irreducible vmem (edge features,
// metadata, h2[src] gathers, global_atomic_add_f32 scatters) + 48 WMMAs.
// ---------------------------------------------------------------------------
__global__ __launch_bounds__(256)
void edge_msg_kernel(const float* __restrict__ ef,
                     const int*   __restrict__ esrc,
                     const int*   __restrict__ edst,
                     const int*   __restrict__ etyp,
                     const float* __restrict__ W5, const float* __restrict__ b5,
                     const float* __restrict__ h2,
                     float* __restrict__ out, int M)
{
    const int lane   = threadIdx.x & 31;
    const int m      = lane & 15;
    const int khalf  = lane >> 4;
    const int gw     = blockIdx.x * 8 + (threadIdx.x >> 5);
    const int nwaves = gridDim.x * 8;
    const int ntiles = M >> 4;                // M % 16 == 0

    // Register-resident W5 fragments and biases (reused across all tiles)
    v2f   bw[3][4][4];                        // [edge_type][ntile][kstep]
    float bias[3][4];
    #pragma unroll
    for (int t = 0; t < 3; ++t) {
        const float* W = W5 + (size_t)t * F_E * HDIM;   // [3][16][64]
        #pragma unroll
        for (int nt = 0; nt < 4; ++nt) {
            const int col = nt * 16 + m;
            bias[t][nt] = b5[t * HDIM + col];
            #pragma unroll
            for (int k = 0; k < 4; ++k) {
                const int kk = 4 * k + 2 * khalf;
                bw[t][nt][k].x = W[kk * HDIM + col];
                bw[t][nt][k].y = W[(kk + 1) * HDIM + col];
            }
        }
    }

    for (int tile = gw; tile < ntiles; tile += nwaves) {
        const int ebase = tile * 16;

        // A fragments: edge_feat[ebase+m][0..15], 4 K-steps of K=4
        v2f a[4];
        const float* erow = ef + (size_t)(ebase + m) * F_E;
        #pragma unroll
        for (int k = 0; k < 4; ++k) {
            const int kk = 4 * k + 2 * khalf;
            a[k].x = erow[kk];
            a[k].y = erow[kk + 1];
        }

        // per-edge metadata for the 8 M-rows this lane's C covers
        int et[8], sv[8], dv[8];
        #pragma unroll
        for (int v = 0; v < 8; ++v) {
            const int e = ebase + v + 8 * khalf;
            et[v] = etyp[e];
            sv[v] = esrc[e];
            dv[v] = edst[e];
        }

        #pragma unroll
        for (int nt = 0; nt < 4; ++nt) {
            const int col = nt * 16 + m;
            v8f acc[3] = { {}, {}, {} };
            #pragma unroll
            for (int t = 0; t < 3; ++t) {
                #pragma unroll
                for (int k = 0; k < 4; ++k)
                    acc[t] = __builtin_amdgcn_wmma_f32_16x16x4_f32(
                                 false, a[k], false, bw[t][nt][k],
                                 (short)0, acc[t], false, false);
            }
            #pragma unroll
            for (int v = 0; v < 8; ++v) {
                const int   t   = et[v];
                const float w5e = (t == 0 ? acc[0][v] : (t == 1 ? acc[1][v] : acc[2][v]))
                                + (t == 0 ? bias[0][nt] : (t == 1 ? bias[1][nt] : bias[2][nt]));
                const float msg = w5e + h2[(size_t)sv[v] * HDIM + col];
                atomicAdd(&out[(size_t)dv[v] * HDIM + col], msg);  // global_atomic_add_f32
            }
        }
    }
}

// ---------------------------------------------------------------------------
// Launch. Input order = setup_inputs() dict order:
//  0:x 1:edge_feat 2:node_type 3:edge_src 4:edge_dst 5:edge_type
//  6:W1 7:b1 8:W2 9:b2 10:W3 11:b3 12:W4 13:b4 14:W5 15:b5
// W3/b3/W4/b4 are dead: softmax over a singleton axis is identically 1.0.
// ---------------------------------------------------------------------------
extern "C" void kernel_launch(void* const* d_in, const int* in_sizes, int n_in,
                              void* d_out, int out_size, void* d_ws, size_t ws_size,
                              hipStream_t stream)
{
    const float* x    = (const float*)d_in[0];
    const float* ef   = (const float*)d_in[1];
    const int*   ntp  = (const int*)  d_in[2];
    const int*   esrc = (const int*)  d_in[3];
    const int*   edst = (const int*)  d_in[4];
    const int*   etyp = (const int*)  d_in[5];
    const float* W1   = (const float*)d_in[6];
    const float* b1   = (const float*)d_in[7];
    const float* W2   = (const float*)d_in[8];
    const float* b2   = (const float*)d_in[9];
    const float* W5   = (const float*)d_in[14];
    const float* b5   = (const float*)d_in[15];

    const int N = in_sizes[2];   // node_type element count
    const int M = in_sizes[3];   // edge_src element count

    float* out = (float*)d_out;
    float* h2  = (float*)d_ws;   // N * 64 floats of scratch

    const size_t lds_bytes = 2u * 2u * 64u * 64u * sizeof(float);  // 64 KB
    (void)hipFuncSetAttribute((const void*)node_proj_kernel,
                              hipFuncAttributeMaxDynamicSharedMemorySize,
                              (int)lds_bytes);

    const int nblocks = (N + 127) / 128;   // 8 waves/block * 16 nodes/wave
    node_proj_kernel<<<nblocks, 256, lds_bytes, stream>>>(x, ntp, W1, b1, W2, b2,
                                                          out, h2, N);

    // Persistent edge kernel: ~8k waves, each handles ~6 tiles, so the
    // register-resident W5 setup is amortized ~6x.
    int eblocks = (M / 16 + 7) / 8;
    if (eblocks > 1024) eblocks = 1024;
    if (eblocks < 1)    eblocks = 1;
    edge_msg_kernel<<<eblocks, 256, 0, stream>>>(ef, esrc, edst, etyp, W5, b5,
                                                 h2, out, M);
}